// SSM_73461120631523
// MI455X (gfx1250) — compile-verified
//
#include <hip/hip_runtime.h>
#include <hip/hip_bf16.h>

// ---------------------------------------------------------------------------
// Mamba-style SSM layer for MI455X (gfx1250), fp32 end-to-end.
//   H=1024, I=2048, N=16, K=4(conv), R=64, B=2, L=1024
// GEMMs use V_WMMA_F32_16X16X4_F32 (exact fp32 matrix op, wave32).
// ---------------------------------------------------------------------------

#define H_DIM 1024
#define I_DIM 2048
#define N_ST  16
#define R_DIM 64
#define B_SZ  2
#define L_SEQ 1024
#define ROWS  (B_SZ * L_SEQ)     // 2048 token rows
#define EPS   1e-06f

typedef __attribute__((ext_vector_type(2))) float v2f;
typedef __attribute__((ext_vector_type(8))) float v8f;

// explicit global-address-space pointers: keeps every fragment load on the
// GLOBAL path (global_load_b64, LOADcnt only) instead of FLAT (LOADcnt+DScnt)
typedef __attribute__((address_space(1))) const float gcf;
typedef __attribute__((address_space(1))) const v2f  gcv2f;

// ---------------------------------------------------------------------------
// Generic fp32 WMMA GEMM:  C[m][n] = sum_k A[m][k] * W[n][k]   (+ epilogue)
//   A: M x K row-major, W: N x K row-major, C: M x N row-major.
//   Wave computes a 16 x (16*NT) tile. Block = 8 waves = 128 rows.
//   M % 128 == 0, N % (16*NT) == 0, K % 16 == 0.
// K-loop: step 16, all (1+NT)*4 v2f fragments loaded into arrays up front
// (single basic block -> backend clauses the loads, partial loadcnt waits
// interleave with the 4*NT independent WMMAs). Prefetch is unconditional
// (speculative; dropped on translation fault per ISA 10.5).
// EPI: 0 = plain store, 1 = softplus(x + bias[n])
// ---------------------------------------------------------------------------
template <int NT, int EPI>
__global__ __launch_bounds__(256)
void wmma_gemm_f32(const float* __restrict__ A,
                   const float* __restrict__ W,
                   const float* __restrict__ bias,
                   float* __restrict__ C,
                   int M, int N, int K)
{
    const int lane = threadIdx.x & 31;
    const int wave = threadIdx.x >> 5;
    const int row0 = (blockIdx.y * 8 + wave) * 16;
    const int col0 = blockIdx.x * (16 * NT);

    const int r  = lane & 15;          // M index (A frag) == N index (B frag / D)
    const int kk = (lane >> 4) * 2;    // K sub-pair selected by half-wave

    gcf* Ap = (gcf*)(A + (size_t)(row0 + r) * K + kk);
    gcf* Wp[NT];
#pragma unroll
    for (int t = 0; t < NT; ++t)
        Wp[t] = (gcf*)(W + (size_t)(col0 + t * 16 + r) * K + kk);

    v8f acc[NT];
#pragma unroll
    for (int t = 0; t < NT; ++t) acc[t] = (v8f){0.f,0.f,0.f,0.f,0.f,0.f,0.f,0.f};

    for (int k = 0; k < K; k += 16) {
        // speculative streaming prefetch of the next A slab
        __builtin_prefetch((const float*)(Ap + 256), 0, 1);

        // ---- load ALL fragments first (one basic block -> load clause) ----
        v2f a[4];
#pragma unroll
        for (int s = 0; s < 4; ++s)
            a[s] = *(gcv2f*)(Ap + 4 * s);

        v2f b[NT][4];
#pragma unroll
        for (int t = 0; t < NT; ++t)
#pragma unroll
            for (int s = 0; s < 4; ++s)
                b[t][s] = *(gcv2f*)(Wp[t] + 4 * s);

        // ---- 4*NT independent WMMAs (K chunks x col tiles) ----
#pragma unroll
        for (int s = 0; s < 4; ++s)
#pragma unroll
            for (int t = 0; t < NT; ++t)
                acc[t] = __builtin_amdgcn_wmma_f32_16x16x4_f32(
                             false, a[s], false, b[t][s],
                             (short)0, acc[t], false, false);

        Ap += 16;
#pragma unroll
        for (int t = 0; t < NT; ++t) Wp[t] += 16;
    }

    // D layout: VGPR v -> row (row0 + v + 8*(lane>=16)), col (col0 + t*16 + r)
    const int rowoff = (lane >> 4) * 8;
#pragma unroll
    for (int t = 0; t < NT; ++t) {
        const int col = col0 + t * 16 + r;
#pragma unroll
        for (int v = 0; v < 8; ++v) {
            float val = acc[t][v];
            if (EPI == 1) {   // softplus(x + bias)
                val += bias[col];
                val = (val > 20.f) ? val : __logf(1.f + __expf(val));
            }
            C[(size_t)(row0 + rowoff + v) * N + col] = val;
        }
    }
}

// ---------------------------------------------------------------------------
// Depthwise causal-ish conv (K=4, pad (1,2)) + bias + SiLU.
//   u[b,l,i] = silu(cb[i] + sum_k cw[i,k] * hs[b, l+k-1, i])
//   hs = proj[..., :I]   (proj row stride = 2*I)
// ---------------------------------------------------------------------------
__global__ __launch_bounds__(256)
void conv_silu_kernel(const float* __restrict__ proj,
                      const float* __restrict__ cw,
                      const float* __restrict__ cb,
                      float* __restrict__ u)
{
    const int row = blockIdx.x;                 // b*L + l
    const int i   = blockIdx.y * 256 + threadIdx.x;
    const int b   = row / L_SEQ;
    const int l   = row - b * L_SEQ;

    float acc = cb[i];
#pragma unroll
    for (int k = 0; k < 4; ++k) {
        const int src = l + k - 1;
        if (src >= 0 && src < L_SEQ)
            acc += cw[i * 4 + k] *
                   proj[(size_t)(b * L_SEQ + src) * (2 * I_DIM) + i];
    }
    u[(size_t)row * I_DIM + i] = acc / (1.f + __expf(-acc));  // SiLU
}

// ---------------------------------------------------------------------------
// Per-row RMSNorm of x_dbl (96 = 64 dt | 16 B | 16 C) with separate weights.
// ---------------------------------------------------------------------------
__global__ __launch_bounds__(256)
void rmsnorm_kernel(const float* __restrict__ xd,
                    const float* __restrict__ wdt,
                    const float* __restrict__ wb,
                    const float* __restrict__ wc,
                    float* __restrict__ dtn,
                    float* __restrict__ Bm,
                    float* __restrict__ Cm)
{
    const int row = blockIdx.x * 256 + threadIdx.x;
    if (row >= ROWS) return;
    const float* x = xd + (size_t)row * (R_DIM + 2 * N_ST);

    float s = 0.f;
#pragma unroll 8
    for (int j = 0; j < R_DIM; ++j) s += x[j] * x[j];
    float inv = __frsqrt_rn(s / R_DIM + EPS);
#pragma unroll 8
    for (int j = 0; j < R_DIM; ++j)
        dtn[(size_t)row * R_DIM + j] = wdt[j] * x[j] * inv;

    s = 0.f;
#pragma unroll
    for (int j = 0; j < N_ST; ++j) { float v = x[R_DIM + j]; s += v * v; }
    inv = __frsqrt_rn(s / N_ST + EPS);
#pragma unroll
    for (int j = 0; j < N_ST; ++j)
        Bm[(size_t)row * N_ST + j] = wb[j] * x[R_DIM + j] * inv;

    s = 0.f;
#pragma unroll
    for (int j = 0; j < N_ST; ++j) { float v = x[R_DIM + N_ST + j]; s += v * v; }
    inv = __frsqrt_rn(s / N_ST + EPS);
#pragma unroll
    for (int j = 0; j < N_ST; ++j)
        Cm[(size_t)row * N_ST + j] = wc[j] * x[R_DIM + N_ST + j] * inv;
}

// ---------------------------------------------------------------------------
// Selective-scan. One thread owns channel (b,i): 16 states in VGPRs,
// serial over L. Bm/Cm (identical across all i of a batch) staged through
// LDS in 64-step chunks and read as broadcasts. Fuses D-skip + gate SiLU.
// ---------------------------------------------------------------------------
#define CHUNK 64
__global__ __launch_bounds__(128)
void scan_kernel(const float* __restrict__ dt2,
                 const float* __restrict__ u,
                 const float* __restrict__ Bm,
                 const float* __restrict__ Cm,
                 const float* __restrict__ A_log,
                 const float* __restrict__ Dvec,
                 const float* __restrict__ proj,
                 float* __restrict__ yout)
{
    __shared__ float sB[CHUNK][N_ST];
    __shared__ float sC[CHUNK][N_ST];

    const int b = blockIdx.y;
    const int i = blockIdx.x * 128 + threadIdx.x;

    float Aa[N_ST];
#pragma unroll
    for (int n = 0; n < N_ST; ++n)
        Aa[n] = -__expf(A_log[(size_t)i * N_ST + n]);   // A = -exp(A_log)

    float h[N_ST];
#pragma unroll
    for (int n = 0; n < N_ST; ++n) h[n] = 0.f;

    const float Dv = Dvec[i];

    for (int l0 = 0; l0 < L_SEQ; l0 += CHUNK) {
        for (int t = threadIdx.x; t < CHUNK * N_ST; t += 128) {
            const size_t g = (size_t)(b * L_SEQ + l0) * N_ST + t;
            sB[t / N_ST][t % N_ST] = Bm[g];
            sC[t / N_ST][t % N_ST] = Cm[g];
        }
        __syncthreads();

        for (int l = 0; l < CHUNK; ++l) {
            const size_t idx = (size_t)(b * L_SEQ + l0 + l) * I_DIM + i;
            const float dtv = dt2[idx];
            const float uv  = u[idx];
            const float du  = dtv * uv;
            float y = 0.f;
#pragma unroll
            for (int n = 0; n < N_ST; ++n) {
                const float dA = __expf(dtv * Aa[n]);   // v_exp_f32 (trans)
                h[n] = dA * h[n] + du * sB[l][n];
                y   += h[n] * sC[l][n];
            }
            const float g = proj[(size_t)(b * L_SEQ + l0 + l) * (2 * I_DIM)
                                 + I_DIM + i];
            yout[idx] = (y + uv * Dv) * (g / (1.f + __expf(-g)));
        }
        __syncthreads();
    }
}

// ---------------------------------------------------------------------------
// Host side: 7 launches, all graph-capture safe on `stream`.
// ---------------------------------------------------------------------------
extern "C" void kernel_launch(void* const* d_in, const int* in_sizes, int n_in,
                              void* d_out, int out_size, void* d_ws, size_t ws_size,
                              hipStream_t stream)
{
    const float* hidden     = (const float*)d_in[0];   // (B,L,H)
    const float* in_proj_w  = (const float*)d_in[1];   // (2I,H)
    const float* conv_w     = (const float*)d_in[2];   // (I,1,4)
    const float* conv_b     = (const float*)d_in[3];   // (I)
    const float* x_proj_w   = (const float*)d_in[4];   // (R+2N, I)
    const float* dt_proj_w  = (const float*)d_in[5];   // (I,R)
    const float* dt_proj_b  = (const float*)d_in[6];   // (I)
    const float* dt_ln_w    = (const float*)d_in[7];   // (R)
    const float* b_ln_w     = (const float*)d_in[8];   // (N)
    const float* c_ln_w     = (const float*)d_in[9];   // (N)
    const float* A_log      = (const float*)d_in[10];  // (I,N)
    const float* Dvec       = (const float*)d_in[11];  // (I)
    const float* out_proj_w = (const float*)d_in[12];  // (H,I)
    float* out = (float*)d_out;

    // workspace carve-up (floats)
    float* ws   = (float*)d_ws;
    float* proj = ws;                                     // ROWS * 2I
    float* u    = proj + (size_t)ROWS * 2 * I_DIM;        // ROWS * I
    float* xdbl = u    + (size_t)ROWS * I_DIM;            // ROWS * 96
    float* dtn  = xdbl + (size_t)ROWS * (R_DIM + 2 * N_ST); // ROWS * 64
    float* Bm   = dtn  + (size_t)ROWS * R_DIM;            // ROWS * 16
    float* Cm   = Bm   + (size_t)ROWS * N_ST;             // ROWS * 16
    float* dt2  = Cm   + (size_t)ROWS * N_ST;             // ROWS * I
    float* yout = dt2  + (size_t)ROWS * I_DIM;            // ROWS * I

    // 1) in_proj: (2048,1024) x (4096,1024)^T -> proj (2048,4096)
    wmma_gemm_f32<4, 0><<<dim3((2 * I_DIM) / 64, ROWS / 128), 256, 0, stream>>>(
        hidden, in_proj_w, nullptr, proj, ROWS, 2 * I_DIM, H_DIM);

    // 2) depthwise conv + SiLU -> u (2048,2048)
    conv_silu_kernel<<<dim3(ROWS, I_DIM / 256), 256, 0, stream>>>(
        proj, conv_w, conv_b, u);

    // 3) x_proj: (2048,2048) x (96,2048)^T -> xdbl (2048,96)
    wmma_gemm_f32<2, 0><<<dim3((R_DIM + 2 * N_ST) / 32, ROWS / 128), 256, 0, stream>>>(
        u, x_proj_w, nullptr, xdbl, ROWS, R_DIM + 2 * N_ST, I_DIM);

    // 4) RMSNorm splits -> dtn, Bm, Cm
    rmsnorm_kernel<<<(ROWS + 255) / 256, 256, 0, stream>>>(
        xdbl, dt_ln_w, b_ln_w, c_ln_w, dtn, Bm, Cm);

    // 5) dt_proj + bias + softplus: (2048,64) x (2048,64)^T -> dt2 (2048,2048)
    wmma_gemm_f32<4, 1><<<dim3(I_DIM / 64, ROWS / 128), 256, 0, stream>>>(
        dtn, dt_proj_w, dt_proj_b, dt2, ROWS, I_DIM, R_DIM);

    // 6) selective scan (+ D skip + gate SiLU) -> yout (2048,2048)
    scan_kernel<<<dim3(I_DIM / 128, B_SZ), 128, 0, stream>>>(
        dt2, u, Bm, Cm, A_log, Dvec, proj, yout);

    // 7) out_proj: (2048,2048) x (1024,2048)^T -> out (2048,1024)
    wmma_gemm_f32<4, 0><<<dim3(H_DIM / 64, ROWS / 128), 256, 0, stream>>>(
        yout, out_proj_w, nullptr, out, ROWS, H_DIM, I_DIM);
}